// GDN_62792421868187
// MI455X (gfx1250) — compile-verified
//
#include <hip/hip_runtime.h>
#include <hip/hip_bf16.h>

typedef __attribute__((ext_vector_type(2))) float v2f;
typedef __attribute__((ext_vector_type(8))) float v8f;

namespace {
constexpr int kB    = 32;
constexpr int kF    = 10;
constexpr int kFp   = 12;             // K padded to multiple of 4 for WMMA
constexpr int kN    = 512;
constexpr int kE    = 8192;
constexpr int kT    = kB * kN;        // 16384 nodes total
constexpr int kH    = 2;
constexpr int kC    = 256;
constexpr int kHC   = kH * kC;        // 512
constexpr int kEtot = kB * kE + kT;   // 278528 edges incl. self loops
constexpr float kSlope = 0.2f;
}

// ---------------------------------------------------------------------------
// Zero-pad layer-1 operands so the GEMM needs only the aligned K path.
//   padA[kT,12]  <- x[kT,10] (cols 10,11 = 0)
//   padW[12,512] <- W1[10,512] (rows 10,11 = 0)
// ---------------------------------------------------------------------------
__global__ void pack_pad(const float* __restrict__ x, float* __restrict__ padA,
                         const float* __restrict__ W1, float* __restrict__ padW) {
  const int i = blockIdx.x * blockDim.x + threadIdx.x;
  if (i < kT * kFp) {
    const int t = i / kFp, f = i - t * kFp;
    padA[i] = (f < kF) ? x[t * kF + f] : 0.f;
  }
  if (i < kFp * kHC) {
    const int r = i / kHC, c = i - r * kHC;
    padW[i] = (r < kF) ? W1[r * kHC + c] : 0.f;
  }
}

// ---------------------------------------------------------------------------
// FP32 WMMA GEMM: C[M,512] = A[M,K] @ B[K,512], row-major, K % 4 == 0.
// Each wave computes a 16x64 tile with 4 v8f accumulators (A frag reused 4x).
// Uses V_WMMA_F32_16X16X4_F32 (exact fp32 math, matches reference precision).
// A frag (16x4): lane<16 -> M=lane, K={kb,kb+1}; lane>=16 -> M=lane-16, K={kb+2,kb+3}
// B frag (4x16): lane<16 -> N=lane, rows {kb,kb+1}; lane>=16 -> rows {kb+2,kb+3}
// C/D (16x16):  VGPR r -> M=row0+r (lanes 0-15) / row0+r+8 (lanes 16-31), N=lane&15
// ldb = ldc = 512 hardcoded so all loads/stores use immediate offsets and the
// K-loop advances by pointer increments only (no in-loop address multiplies).
// ---------------------------------------------------------------------------
__global__ void wmma_gemm_f32(const float* __restrict__ A, int lda,
                              const float* __restrict__ Bm,
                              float* __restrict__ C,
                              int M, int K) {
  constexpr int kLdb = kHC;                     // 512
  const int wid    = (blockIdx.x * blockDim.x + threadIdx.x) >> 5;
  const int lane   = threadIdx.x & 31;
  const int wavesN = kHC >> 6;                  // 8 column tiles of 64
  const int mt     = wid / wavesN;
  const int nt     = wid - mt * wavesN;
  const int row0   = mt << 4;
  const int col0   = nt << 6;
  if (row0 >= M) return;                        // uniform per wave (EXEC stays all-1)

  const int half = lane >> 4;                   // which K-pair this lane owns
  const int l16  = lane & 15;

  const float* Ap = A  + (size_t)(row0 + l16) * lda + (half << 1);
  const float* Bp = Bm + (size_t)(half << 1) * kLdb + col0 + l16;

  v8f acc0 = {}, acc1 = {}, acc2 = {}, acc3 = {};

#pragma unroll 2
  for (int kb = 0; kb < K; kb += 4) {
    const v2f a = *(const v2f*)Ap;              // 8B load, provably aligned
    v2f b0, b1, b2, b3;
    b0.x = Bp[0];          b0.y = Bp[kLdb + 0];
    b1.x = Bp[16];         b1.y = Bp[kLdb + 16];
    b2.x = Bp[32];         b2.y = Bp[kLdb + 32];
    b3.x = Bp[48];         b3.y = Bp[kLdb + 48];
    acc0 = __builtin_amdgcn_wmma_f32_16x16x4_f32(false, a, false, b0, (short)0, acc0, false, false);
    acc1 = __builtin_amdgcn_wmma_f32_16x16x4_f32(false, a, false, b1, (short)0, acc1, false, false);
    acc2 = __builtin_amdgcn_wmma_f32_16x16x4_f32(false, a, false, b2, (short)0, acc2, false, false);
    acc3 = __builtin_amdgcn_wmma_f32_16x16x4_f32(false, a, false, b3, (short)0, acc3, false, false);
    Ap += 4;
    Bp += 4 * kLdb;
  }

  float* Cp = C + (size_t)(row0 + (half << 3)) * kHC + col0 + l16;
#pragma unroll
  for (int r = 0; r < 8; ++r) {
    Cp[0]  = acc0[r];
    Cp[16] = acc1[r];
    Cp[32] = acc2[r];
    Cp[48] = acc3[r];
    Cp += kHC;
  }
}

// ---------------------------------------------------------------------------
// Per-(node,head) attention logits: als = <h, a_src>, ald = <h, a_dst>.
// One wave per (node,head); shuffle reduction.
// ---------------------------------------------------------------------------
__global__ void node_alpha(const float* __restrict__ h,
                           const float* __restrict__ a_s,
                           const float* __restrict__ a_d,
                           float* __restrict__ als, float* __restrict__ ald) {
  const int wid  = (blockIdx.x * blockDim.x + threadIdx.x) >> 5;
  const int lane = threadIdx.x & 31;
  if (wid >= kT * kH) return;
  const int t = wid >> 1, hh = wid & 1;
  const float* hp = h + (size_t)t * kHC + hh * kC;
  const float* ap = a_s + hh * kC;
  const float* dp = a_d + hh * kC;
  float s = 0.f, d = 0.f;
#pragma unroll
  for (int c = lane; c < kC; c += 32) {
    const float v = hp[c];
    s += v * ap[c];
    d += v * dp[c];
  }
#pragma unroll
  for (int off = 16; off; off >>= 1) {
    s += __shfl_down(s, off, 32);
    d += __shfl_down(d, off, 32);
  }
  if (lane == 0) { als[wid] = s; ald[wid] = d; }
}

// ---------------------------------------------------------------------------
// Init per-layer state: out := bias (accumulator seed), emax := -inf, denom := 0
// ---------------------------------------------------------------------------
__global__ void init_layer(float* __restrict__ out, const float* __restrict__ bias,
                           float* __restrict__ emax, float* __restrict__ denom) {
  const int i = blockIdx.x * blockDim.x + threadIdx.x;
  if (i < kT * kHC) out[i] = bias[i & (kHC - 1)];
  if (i < kT * kH) { emax[i] = -__builtin_inff(); denom[i] = 0.f; }
}

__device__ __forceinline__ void edge_nodes(int g, const int* __restrict__ es,
                                           const int* __restrict__ ed,
                                           int& s, int& d) {
  if (g < kB * kE) {
    const int b = g >> 13;                 // g / kE (kE = 8192)
    const int k = g & (kE - 1);
    s = es[k] + b * kN;
    d = ed[k] + b * kN;
  } else {
    s = d = g - kB * kE;                   // self loop
  }
}

__device__ __forceinline__ float lrelu(float x) { return x > 0.f ? x : kSlope * x; }

__device__ __forceinline__ void atomic_max_f32(float* addr, float v) {
  // valid for mixed signs given -inf initialization
  if (v >= 0.f) atomicMax((int*)addr, __float_as_int(v));
  else          atomicMin((unsigned int*)addr, __float_as_uint(v));
}

// ---------------------------------------------------------------------------
// Segment softmax pass 1: per-dst max of leaky_relu(als[src] + ald[dst])
// ---------------------------------------------------------------------------
__global__ void edge_max(const int* __restrict__ es, const int* __restrict__ ed,
                         const float* __restrict__ als, const float* __restrict__ ald,
                         float* __restrict__ emax) {
  const int g = blockIdx.x * blockDim.x + threadIdx.x;
  if (g >= kEtot) return;
  int s, d;
  edge_nodes(g, es, ed, s, d);
#pragma unroll
  for (int hh = 0; hh < kH; ++hh) {
    const float e = lrelu(als[s * kH + hh] + ald[d * kH + hh]);
    atomic_max_f32(&emax[d * kH + hh], e);
  }
}

// ---------------------------------------------------------------------------
// Segment softmax pass 2: ex = exp(e - emax[dst]); denom[dst] += ex
// ---------------------------------------------------------------------------
__global__ void edge_exp(const int* __restrict__ es, const int* __restrict__ ed,
                         const float* __restrict__ als, const float* __restrict__ ald,
                         const float* __restrict__ emax,
                         float* __restrict__ exb, float* __restrict__ denom) {
  const int g = blockIdx.x * blockDim.x + threadIdx.x;
  if (g >= kEtot) return;
  int s, d;
  edge_nodes(g, es, ed, s, d);
#pragma unroll
  for (int hh = 0; hh < kH; ++hh) {
    const float e  = lrelu(als[s * kH + hh] + ald[d * kH + hh]);
    const float ex = expf(e - emax[d * kH + hh]);
    exb[g * kH + hh] = ex;
    atomicAdd(&denom[d * kH + hh], ex);
  }
}

// ---------------------------------------------------------------------------
// Message scatter: one wave per (edge, head); 32-lane coalesced atomic adds.
// out[dst, hh, :] += alpha * h[src, hh, :]
// ---------------------------------------------------------------------------
__global__ void edge_scatter(const int* __restrict__ es, const int* __restrict__ ed,
                             const float* __restrict__ h,
                             const float* __restrict__ exb,
                             const float* __restrict__ denom,
                             float* __restrict__ out) {
  const int wid  = (blockIdx.x * blockDim.x + threadIdx.x) >> 5;
  const int lane = threadIdx.x & 31;
  if (wid >= kEtot * kH) return;
  const int g = wid >> 1, hh = wid & 1;
  int s, d;
  edge_nodes(g, es, ed, s, d);
  const float alpha = exb[wid] / (denom[d * kH + hh] + 1e-16f);
  const float* hp = h + (size_t)s * kHC + hh * kC;
  float* op = out + (size_t)d * kHC + hh * kC;
#pragma unroll
  for (int c = lane; c < kC; c += 32)
    atomicAdd(&op[c], hp[c] * alpha);
}

// ---------------------------------------------------------------------------
// Final: y = relu( (x2 @ Wfc + bfc) * gamma/sqrt(1+eps) + beta ), wave per node
// ---------------------------------------------------------------------------
__global__ void final_fc(const float* __restrict__ x2, const float* __restrict__ Wfc,
                         const float* __restrict__ bfc, const float* __restrict__ gamma,
                         const float* __restrict__ beta, float* __restrict__ out) {
  const int wid  = (blockIdx.x * blockDim.x + threadIdx.x) >> 5;
  const int lane = threadIdx.x & 31;
  if (wid >= kT) return;
  const float* xp = x2 + (size_t)wid * kHC;
  float s = 0.f;
#pragma unroll
  for (int j = lane; j < kHC; j += 32) s += xp[j] * Wfc[j];
#pragma unroll
  for (int off = 16; off; off >>= 1) s += __shfl_down(s, off, 32);
  if (lane == 0) {
    float y = s + bfc[0];
    y = y * (gamma[0] * rsqrtf(1.f + 1e-5f)) + beta[0];
    out[wid] = fmaxf(y, 0.f);
  }
}

extern "C" void kernel_launch(void* const* d_in, const int* in_sizes, int n_in,
                              void* d_out, int out_size, void* d_ws, size_t ws_size,
                              hipStream_t stream) {
  const float* data  = (const float*)d_in[0];   // [B,F,N] -> viewed as [T, 10] row-major
  const int*   eidx  = (const int*)d_in[1];     // [2, E]
  const float* W1    = (const float*)d_in[2];
  const float* as1   = (const float*)d_in[3];
  const float* ad1   = (const float*)d_in[4];
  const float* b1    = (const float*)d_in[5];
  const float* W2    = (const float*)d_in[6];
  const float* as2   = (const float*)d_in[7];
  const float* ad2   = (const float*)d_in[8];
  const float* b2    = (const float*)d_in[9];
  const float* Wfc   = (const float*)d_in[10];
  const float* bfc   = (const float*)d_in[11];
  const float* gamma = (const float*)d_in[12];
  const float* beta  = (const float*)d_in[13];
  float* out = (float*)d_out;

  // workspace carve-out (~71 MB, all L2-resident)
  float* ws    = (float*)d_ws;
  float* bufH  = ws;                          // [kT, kHC] h (per layer)
  float* bufX  = bufH + (size_t)kT * kHC;     // [kT, kHC] x1, later x2
  float* als   = bufX + (size_t)kT * kHC;     // [kT*kH]
  float* ald   = als + kT * kH;
  float* emax  = ald + kT * kH;
  float* denom = emax + kT * kH;
  float* exb   = denom + kT * kH;             // [kEtot*kH]
  float* padA  = exb + (size_t)kEtot * kH;    // [kT, 12]
  float* padW  = padA + (size_t)kT * kFp;     // [12, 512]

  const int* es = eidx;                        // src row
  const int* ed = eidx + kE;                   // dst row

  const int gemm_waves   = (kT / 16) * (kHC / 64);          // 8192
  const int gemm_blocks  = gemm_waves * 32 / 256;           // 1024
  const int alpha_blocks = kT * kH * 32 / 256;              // 4096
  const int init_blocks  = kT * kHC / 256;                  // 32768
  const int edge_blocks  = (kEtot + 255) / 256;             // 1088
  const int scat_blocks  = kEtot * kH * 32 / 256;           // 69632
  const int fc_blocks    = kT * 32 / 256;                   // 2048
  const int pad_blocks   = (kT * kFp + 255) / 256;          // 768

  // ---- layer 1: pad K 10 -> 12, then aligned WMMA GEMM ----
  pack_pad<<<pad_blocks, 256, 0, stream>>>(data, padA, W1, padW);
  wmma_gemm_f32<<<gemm_blocks, 256, 0, stream>>>(padA, kFp, padW, bufH, kT, kFp);
  node_alpha<<<alpha_blocks, 256, 0, stream>>>(bufH, as1, ad1, als, ald);
  init_layer<<<init_blocks, 256, 0, stream>>>(bufX, b1, emax, denom);
  edge_max<<<edge_blocks, 256, 0, stream>>>(es, ed, als, ald, emax);
  edge_exp<<<edge_blocks, 256, 0, stream>>>(es, ed, als, ald, emax, exb, denom);
  edge_scatter<<<scat_blocks, 256, 0, stream>>>(es, ed, bufH, exb, denom, bufX);

  // ---- layer 2: h = x1 @ W2 (K=512) ----
  wmma_gemm_f32<<<gemm_blocks, 256, 0, stream>>>(bufX, kHC, W2, bufH, kT, kHC);
  node_alpha<<<alpha_blocks, 256, 0, stream>>>(bufH, as2, ad2, als, ald);
  init_layer<<<init_blocks, 256, 0, stream>>>(bufX, b2, emax, denom);  // x1 consumed above
  edge_max<<<edge_blocks, 256, 0, stream>>>(es, ed, als, ald, emax);
  edge_exp<<<edge_blocks, 256, 0, stream>>>(es, ed, als, ald, emax, exb, denom);
  edge_scatter<<<scat_blocks, 256, 0, stream>>>(es, ed, bufH, exb, denom, bufX);

  // ---- head: fc + BN(eval) + relu ----
  final_fc<<<fc_blocks, 256, 0, stream>>>(bufX, Wfc, bfc, gamma, beta, out);
}